// CTCLayer_35192962023795
// MI455X (gfx1250) — compile-verified
//
#include <hip/hip_runtime.h>
#include <cstdint>

// ---------------------------------------------------------------------------
// CTC forward loss for MI455X (gfx1250).
//   B=512, T=512, C=96, L=64  ->  Tp=510, S=129, BLANK=95
// Bandwidth-bound (~100 MB of y_pred) with a serial-in-t DP per batch element.
// One wave32 per batch element; alpha register-resident (5 states/lane).
// y_pred rows are double-buffered into LDS with gfx1250 async global->LDS
// copies (ASYNCcnt pipeline) so streaming overlaps the DP math.
// DP runs in SCALED LINEAR space (renormalize by the state-sum each step,
// accumulate log-scale in fp64): exact vs. the log-space reference, but only
// 1 v_log_f32 + 2 v_rcp_f32 per timestep instead of ~26 transcendentals.
// ---------------------------------------------------------------------------

namespace {
constexpr int kB = 512;
constexpr int kT = 512;
constexpr int kC = 96;
constexpr int kL = 64;
constexpr int kBlank = kC - 1;           // 95
constexpr int kTp = kT - 2;              // 510
constexpr int kS  = 2 * kL + 1;          // 129
constexpr float kEps = 1e-7f;
constexpr int kChunk = 32;                              // timesteps per LDS chunk
constexpr int kNChunk = (kTp + kChunk - 1) / kChunk;    // 16
constexpr int kF4PerChunk = kChunk * kC / 4;            // 768 float4 per chunk
constexpr int kF4PerLane  = kF4PerChunk / 32;           // 24 async ops / lane / chunk
}  // namespace

template <int N>
__device__ __forceinline__ void wait_asynccnt() {
#if defined(__has_builtin) && __has_builtin(__builtin_amdgcn_s_wait_asynccnt)
  __builtin_amdgcn_s_wait_asynccnt(N);
#else
  asm volatile("s_wait_asynccnt %0" ::"n"(N) : "memory");
#endif
}

// gfx1250 async copy engine: 16B global -> LDS per lane, tracked by ASYNCcnt.
__device__ __forceinline__ void async_b128(uint32_t lds_off, const void* gaddr) {
  asm volatile("global_load_async_to_lds_b128 %0, %1, off"
               ::"v"(lds_off), "v"(gaddr)
               : "memory");
}

// Copy one 32-timestep chunk of y_pred rows for batch b into LDS.
// CLAMP is needed only for the final chunk (rows 482..513 run past row 511).
template <bool CLAMP>
__device__ __forceinline__ void prefetch_chunk(const float* __restrict__ y_pred,
                                               int b, int c, uint32_t lbase,
                                               int lane) {
  const long base4 = ((long)(b * kT + 2 + c * kChunk)) * (kC / 4);
  const long TOT4 = (long)kB * kT * (kC / 4);
#pragma unroll
  for (int i = 0; i < kF4PerLane; ++i) {
    long f4 = base4 + lane + (i << 5);
    if (CLAMP && f4 >= TOT4) f4 = TOT4 - 1;   // in-bounds dummy read (unused)
    const void* g = (const char*)y_pred + (f4 << 4);
    uint32_t l = lbase + ((uint32_t)(lane + (i << 5)) << 4);
    async_b128(l, g);
  }
}

__global__ __launch_bounds__(32) void ctc_fwd_kernel(
    const int* __restrict__ y_true,        // [B, L]
    const float* __restrict__ y_pred,      // [B, T, C] softmax probs
    const int* __restrict__ input_len,     // [B]
    const int* __restrict__ label_len,     // [B]
    float* __restrict__ out) {             // [B]
  __shared__ float buf[2][kChunk * kC];    // double-buffered prob rows (24 KB)
  __shared__ int   slab[kL];
  __shared__ float afin[160];

  const int b = blockIdx.x;
  const int lane = threadIdx.x;

  slab[lane]      = y_true[b * kL + lane];
  slab[lane + 32] = y_true[b * kL + lane + 32];

  const int ilen = input_len[b];
  const int llen = label_len[b];

  // LDS byte offsets (low 32 bits of an LDS object's flat address).
  const uint32_t lds0 = (uint32_t)(uintptr_t)&buf[0][0];
  const uint32_t lds1 = (uint32_t)(uintptr_t)&buf[1][0];

  prefetch_chunk<false>(y_pred, b, 0, lds0, lane);   // ASYNCcnt: 24
  prefetch_chunk<false>(y_pred, b, 1, lds1, lane);   // ASYNCcnt: 48

  __syncthreads();  // single-wave WG -> S_NOP; orders slab writes/reads

  // Per-lane extended-label metadata: lane owns states s = 5*lane + j.
  // States s >= 129 only receive mass from below and are never read: they
  // perturb the (arbitrary) per-step scale only, which cancels exactly in L.
  int  e[5];
  bool sk[5];
#pragma unroll
  for (int j = 0; j < 5; ++j) {
    int s = 5 * lane + j;
    if (s < kS) {
      e[j]  = (s & 1) ? slab[s >> 1] : kBlank;
      sk[j] = (s & 1) && (s >= 2) && (slab[s >> 1] != slab[(s >> 1) - 1]);
    } else {
      e[j] = kBlank;
      sk[j] = false;
    }
  }

  wait_asynccnt<kF4PerLane>();  // <=24 outstanding -> chunk 0 resident

  // t = 0:  alpha[s] = (s < 2) ? (p_ext + eps)/denom : 0, in scaled form.
  float a[5];
  double L;  // accumulated log-scale: true alpha = exp(L) * a
  {
    const float* row = &buf[0][0];
    float pd = row[lane] + row[lane + 32] + row[lane + 64];
#pragma unroll
    for (int o = 16; o > 0; o >>= 1) pd += __shfl_xor(pd, o);
    const float denom = pd + (float)kC * kEps;

#pragma unroll
    for (int j = 0; j < 5; ++j) {
      int s = 5 * lane + j;
      a[j] = (s < 2) ? (row[e[j]] + kEps) : 0.0f;
    }
    float sg = ((a[0] + a[1]) + (a[2] + a[3])) + a[4];
#pragma unroll
    for (int o = 16; o > 0; o >>= 1) sg += __shfl_xor(sg, o);
    float inv = __builtin_amdgcn_rcpf(sg);
#pragma unroll
    for (int j = 0; j < 5; ++j) a[j] *= inv;
    L = (double)__logf(sg * __builtin_amdgcn_rcpf(denom));
  }

  for (int t = 1; t < kTp; ++t) {
    const int c = t >> 5, tin = t & (kChunk - 1);
    if (tin == 0) {
      if (c + 1 < kNChunk) {
        if (c + 1 == kNChunk - 1)
          prefetch_chunk<true>(y_pred, b, c + 1, ((c + 1) & 1) ? lds1 : lds0, lane);
        else
          prefetch_chunk<false>(y_pred, b, c + 1, ((c + 1) & 1) ? lds1 : lds0, lane);
        wait_asynccnt<kF4PerLane>();   // chunk c now fully resident
      } else {
        wait_asynccnt<0>();            // drain: last chunk resident
      }
    }
    const float* row = &buf[c & 1][tin * kC];

    if (t < ilen) {
      // log-softmax denominator for this row (enters only the scale L)
      float pd = row[lane] + row[lane + 32] + row[lane + 64];
#pragma unroll
      for (int o = 16; o > 0; o >>= 1) pd += __shfl_xor(pd, o);
      const float denom = pd + (float)kC * kEps;

      // neighbor states from the previous lane (its slots 3 and 4)
      float up4 = __shfl_up(a[4], 1);   // state 5*lane - 1
      float up3 = __shfl_up(a[3], 1);   // state 5*lane - 2
      if (lane == 0) { up4 = 0.0f; up3 = 0.0f; }

      float u0 = (a[0] + up4  + (sk[0] ? up3  : 0.0f)) * (row[e[0]] + kEps);
      float u1 = (a[1] + a[0] + (sk[1] ? up4  : 0.0f)) * (row[e[1]] + kEps);
      float u2 = (a[2] + a[1] + (sk[2] ? a[0] : 0.0f)) * (row[e[2]] + kEps);
      float u3 = (a[3] + a[2] + (sk[3] ? a[1] : 0.0f)) * (row[e[3]] + kEps);
      float u4 = (a[4] + a[3] + (sk[4] ? a[2] : 0.0f)) * (row[e[4]] + kEps);

      float sg = ((u0 + u1) + (u2 + u3)) + u4;
#pragma unroll
      for (int o = 16; o > 0; o >>= 1) sg += __shfl_xor(sg, o);

      float inv = __builtin_amdgcn_rcpf(sg);
      a[0] = u0 * inv; a[1] = u1 * inv; a[2] = u2 * inv;
      a[3] = u3 * inv; a[4] = u4 * inv;
      L += (double)__logf(sg * __builtin_amdgcn_rcpf(denom));
    }
  }

  // Stage final scaled alpha and reduce to the loss.
#pragma unroll
  for (int j = 0; j < 5; ++j) afin[5 * lane + j] = a[j];
  __syncthreads();
  if (lane == 0) {
    int sl = 2 * llen;  // <= 128
    float s2 = afin[sl] + afin[sl - 1];
    out[b] = (float)(-(L + (double)__logf(s2)));
  }
}

extern "C" void kernel_launch(void* const* d_in, const int* in_sizes, int n_in,
                              void* d_out, int out_size, void* d_ws, size_t ws_size,
                              hipStream_t stream) {
  (void)in_sizes; (void)n_in; (void)out_size; (void)d_ws; (void)ws_size;
  const int*   y_true = (const int*)d_in[0];
  const float* y_pred = (const float*)d_in[1];
  const int*   il     = (const int*)d_in[2];
  const int*   ll     = (const int*)d_in[3];
  float* out = (float*)d_out;
  ctc_fwd_kernel<<<kB, 32, 0, stream>>>(y_true, y_pred, il, ll, out);
}